// TTD_TransE_TYPE_2_59519656788294
// MI455X (gfx1250) — compile-verified
//
#include <hip/hip_runtime.h>
#include <hip/hip_bf16.h>

typedef float v2f __attribute__((ext_vector_type(2)));
typedef float v8f __attribute__((ext_vector_type(8)));

#ifndef __has_builtin
#define __has_builtin(x) 0
#endif

#if __has_builtin(__builtin_amdgcn_wmma_f32_16x16x4_f32)
#define TTD_USE_WMMA 1
#else
#define TTD_USE_WMMA 0
#endif

#define EPS_NORM 1e-12f
#define EPS_PD   1e-6f

// Score uses only he[0..2]  =>  only rows 0..2 of Mh are needed.
// WMMA path: each wave computes, for 16 samples and row i,
//   C_i[m][n] = sum_j Mh_m[i][j] * h_n[j]   (chained f32 16x16x4 WMMA over K)
// and extracts the diagonal C_i[s][s] via LDS.
__global__ __launch_bounds__(256) void ttd_transe_type_score(
    const int*   __restrict__ sample,      // (B,3) int32: head, rel, tail(unused)
    const float* __restrict__ entity_emb,  // (E,128)
    const float* __restrict__ type_emb,    // (T*R, 128*128)
    const int*   __restrict__ node_type,   // (E,)
    float*       __restrict__ out,         // (B,)
    const int B, const int t_len)
{
#if TTD_USE_WMMA
    __shared__ float cbuf[8][3][16][16];   // [wave][row i][m][n], 24 KB

    const int lane = threadIdx.x & 31;
    const int wave = threadIdx.x >> 5;
    const int s    = lane & 15;            // sample slot within wave (M and N index)
    const int hi   = lane >> 4;            // 0 -> K={0,1}, 1 -> K={2,3} (A/B f32 layout)

    const int gwave = blockIdx.x * (blockDim.x >> 5) + wave;
    int gs = gwave * 16 + s;
    const bool valid = gs < B;
    if (!valid) gs = 0;                    // clamp for safe (discarded) loads

    const int head = sample[gs * 3 + 0];
    const int rel  = sample[gs * 3 + 1];
    const int nt   = node_type[head];

    const size_t matOff = (size_t)(rel * t_len + nt) * (size_t)(128 * 128);
    const int    koff   = hi * 2;

    const float* hp = entity_emb + (size_t)head * 128 + koff;  // B operand source
    const float* r0 = type_emb + matOff + koff;                // A: row 0
    const float* r1 = r0 + 128;                                // A: row 1
    const float* r2 = r0 + 256;                                // A: row 2

    v8f acc0 = {}; v8f acc1 = {}; v8f acc2 = {};

#pragma unroll
    for (int kc = 0; kc < 32; ++kc) {
        const int j = kc * 4;
        v2f bv = *(const v2f*)(hp + j);    // B[k][n] = h_n[4kc+k]
        v2f a0 = *(const v2f*)(r0 + j);    // A[m][k] = Mh_m[i][4kc+k]
        v2f a1 = *(const v2f*)(r1 + j);
        v2f a2 = *(const v2f*)(r2 + j);
        acc0 = __builtin_amdgcn_wmma_f32_16x16x4_f32(false, a0, false, bv, (short)0, acc0, false, false);
        acc1 = __builtin_amdgcn_wmma_f32_16x16x4_f32(false, a1, false, bv, (short)0, acc1, false, false);
        acc2 = __builtin_amdgcn_wmma_f32_16x16x4_f32(false, a2, false, bv, (short)0, acc2, false, false);
    }

    // C/D layout: VGPR j, lanes 0-15 -> M=j, N=lane; lanes 16-31 -> M=j+8, N=lane-16.
#pragma unroll
    for (int jj = 0; jj < 8; ++jj) {
        const int m = jj + hi * 8;
        cbuf[wave][0][m][s] = acc0[jj];
        cbuf[wave][1][m][s] = acc1[jj];
        cbuf[wave][2][m][s] = acc2[jj];
    }
    __syncthreads();

    if (valid && hi == 0) {
        const float d0 = cbuf[wave][0][s][s];
        const float d1 = cbuf[wave][1][s][s];
        const float d2 = cbuf[wave][2][s][s];
        const float n0 = d0 / fmaxf(fabsf(d0), EPS_NORM);
        const float n1 = d1 / fmaxf(fabsf(d1), EPS_NORM);
        const float n2 = d2 / fmaxf(fabsf(d2), EPS_NORM);
        out[gs] = fabsf(n0 + n1 - n2 + EPS_PD);
    }
#else
    // Fallback: one wave per sample, f32 dot3 + wave32 shuffle reduction.
    const int lane = threadIdx.x & 31;
    const int wid  = (blockIdx.x * blockDim.x + threadIdx.x) >> 5;
    const int nw   = (gridDim.x * blockDim.x) >> 5;

    for (int gs = wid; gs < B; gs += nw) {
        const int head = sample[gs * 3 + 0];
        const int rel  = sample[gs * 3 + 1];
        const int nt   = node_type[head];
        const size_t matOff = (size_t)(rel * t_len + nt) * (size_t)(128 * 128);

        const float4 hv = *(const float4*)(entity_emb + (size_t)head * 128 + lane * 4);
        const float4 m0 = *(const float4*)(type_emb + matOff + 0 * 128 + lane * 4);
        const float4 m1 = *(const float4*)(type_emb + matOff + 1 * 128 + lane * 4);
        const float4 m2 = *(const float4*)(type_emb + matOff + 2 * 128 + lane * 4);

        float p0 = m0.x * hv.x + m0.y * hv.y + m0.z * hv.z + m0.w * hv.w;
        float p1 = m1.x * hv.x + m1.y * hv.y + m1.z * hv.z + m1.w * hv.w;
        float p2 = m2.x * hv.x + m2.y * hv.y + m2.z * hv.z + m2.w * hv.w;
#pragma unroll
        for (int off = 16; off > 0; off >>= 1) {
            p0 += __shfl_down(p0, off, 32);
            p1 += __shfl_down(p1, off, 32);
            p2 += __shfl_down(p2, off, 32);
        }
        if (lane == 0) {
            const float n0 = p0 / fmaxf(fabsf(p0), EPS_NORM);
            const float n1 = p1 / fmaxf(fabsf(p1), EPS_NORM);
            const float n2 = p2 / fmaxf(fabsf(p2), EPS_NORM);
            out[gs] = fabsf(n0 + n1 - n2 + EPS_PD);
        }
    }
#endif
}

extern "C" void kernel_launch(void* const* d_in, const int* in_sizes, int n_in,
                              void* d_out, int out_size, void* d_ws, size_t ws_size,
                              hipStream_t stream) {
    (void)n_in; (void)d_ws; (void)ws_size;
    const int*   sample   = (const int*)d_in[0];     // (B,3) int32
    const float* entity   = (const float*)d_in[1];   // (E,128) f32
    // d_in[2] = relation_emb: only its row count matters (t_len derivation)
    const float* type_emb = (const float*)d_in[3];   // (T*R, 128*128) f32
    const int*   ntype    = (const int*)d_in[4];     // (E,) int32

    const int D        = 128;
    const int B        = out_size;
    const int R        = in_sizes[2] / D;
    const int typeRows = (int)((long long)in_sizes[3] / (D * D));
    const int t_len    = (R > 0) ? (typeRows / R) : 16;

    // 256 threads = 8 waves; WMMA path: 16 samples/wave -> 128 samples/block.
    const int blocks = (B + 127) / 128;
    ttd_transe_type_score<<<blocks, 256, 0, stream>>>(
        sample, entity, type_emb, ntype, (float*)d_out, B, t_len);
}